// BiDAF_11132555231298
// MI455X (gfx1250) — compile-verified
//
#include <hip/hip_runtime.h>
#include <hip/hip_bf16.h>
#include <math.h>

// ---------------- model constants ----------------
#define BQ    32
#define CLEN  512
#define QLEN  48
#define WLEN  16
#define HH    100
#define D2    200     // 2*H
#define G8    800     // 8*H
#define MC    (BQ*CLEN)   // 16384 context rows
#define MQ    (BQ*QLEN)   // 1536 query rows

typedef float v2f __attribute__((ext_vector_type(2)));
typedef float v8f __attribute__((ext_vector_type(8)));

// =====================================================================
// WMMA f32 GEMM:  C[M,N] = act( A[M,K] * W[N,K]^T + bias[N] )
// One wave computes a 16x16 tile of C with V_WMMA_F32_16X16X4_F32.
// A-frag (16x4, MxK): lane l -> row m=l%16, k-pair base 2*(l>>4)
// B-frag (4x16, KxN): lane l -> col n=l%16, k-pair base 2*(l>>4)
//   (B[k][n] == W[n][k], so both frags use identical float2 addressing)
// C/D   : lane l holds C[m0+8*(l>>4)+r][n0+l%16] in acc element r.
// act: 0 = none, 1 = relu, 2 = sigmoid
// Requires: K % 4 == 0, M % 16 == 0 (true for all uses here).
// =====================================================================
__global__ void gemm_wmma_f32(const float* __restrict__ A,
                              const float* __restrict__ W,
                              const float* __restrict__ bias,
                              float* __restrict__ C,
                              int M, int N, int K, int act) {
    const int lane = threadIdx.x;              // 0..31
    const int mtile = blockIdx.y * blockDim.y + threadIdx.y;
    const int m0 = mtile * 16;
    const int n0 = blockIdx.x * 16;
    if (m0 >= M) return;                       // wave-uniform: EXEC stays all-1
    const int r16 = lane & 15;
    const int hi  = lane >> 4;                 // 0 or 1
    const int arow = m0 + r16;
    int wrow = n0 + r16; if (wrow >= N) wrow = N - 1;   // clamp, masked at store
    const float* Ap = A + (size_t)arow * K;
    const float* Wp = W + (size_t)wrow * K;

    v8f acc = {0.f,0.f,0.f,0.f,0.f,0.f,0.f,0.f};
    for (int kb = 0; kb < K; kb += 4) {
        v2f a = *(const v2f*)(Ap + kb + 2*hi);
        v2f b = *(const v2f*)(Wp + kb + 2*hi);
        acc = __builtin_amdgcn_wmma_f32_16x16x4_f32(
            /*neg_a=*/false, a, /*neg_b=*/false, b,
            /*c_mod=*/(short)0, acc, /*reuse_a=*/false, /*reuse_b=*/false);
    }

    const int col = n0 + r16;
    if (col < N) {
        const float bv = bias ? bias[col] : 0.0f;
        union { v8f v; float f[8]; } u; u.v = acc;
        #pragma unroll
        for (int r = 0; r < 8; ++r) {
            float v = u.f[r] + bv;
            if (act == 1)      v = v > 0.f ? v : 0.f;
            else if (act == 2) v = 1.0f / (1.0f + __expf(-v));
            C[(size_t)(m0 + 8*hi + r) * N + col] = v;
        }
    }
}

// =====================================================================
// Char CNN: embed 16 chars (dim 8), conv width 5 -> 100 ch, relu, maxpool.
// One block per word; writes out[word*200 + o] (char half of highway input).
// relu(max(conv)) == max(relu(conv)) since relu is monotone.
// =====================================================================
__global__ void char_conv_kernel(const int* __restrict__ chars,
                                 const float* __restrict__ embW,
                                 const float* __restrict__ convW,
                                 const float* __restrict__ convb,
                                 float* __restrict__ out) {
    __shared__ float e[8][16];        // [char_dim][pos]
    const int bl  = blockIdx.x;
    const int tid = threadIdx.x;      // 128 threads
    if (tid < 128) {
        const int w = tid >> 3, i = tid & 7;
        const int idx = chars[(size_t)bl * WLEN + w];
        e[i][w] = embW[(size_t)idx * 8 + i];
    }
    __syncthreads();
    if (tid < 100) {
        const int o = tid;
        float best = -1e30f;
        for (int pos = 0; pos < 12; ++pos) {
            float acc = convb[o];
            #pragma unroll
            for (int i = 0; i < 8; ++i)
                #pragma unroll
                for (int w = 0; w < 5; ++w)
                    acc += e[i][pos + w] * convW[((size_t)o * 8 + i) * 5 + w];
            best = fmaxf(best, acc);
        }
        out[(size_t)bl * D2 + o] = fmaxf(best, 0.0f);
    }
}

// word embedding gather -> word half (cols 100..199) of highway input
__global__ void word_gather_kernel(const int* __restrict__ words,
                                   const float* __restrict__ embW,
                                   float* __restrict__ out, int nWords) {
    const int t = blockIdx.x * blockDim.x + threadIdx.x;
    if (t >= nWords * 100) return;
    const int w = t / 100, d = t % 100;
    out[(size_t)w * D2 + 100 + d] = embW[(size_t)words[w] * 100 + d];
}

// x = g*h + (1-g)*x
__global__ void highway_combine_kernel(float* __restrict__ x,
                                       const float* __restrict__ h,
                                       const float* __restrict__ g, int n) {
    const int t = blockIdx.x * blockDim.x + threadIdx.x;
    if (t >= n) return;
    const float gv = g[t];
    x[t] = gv * h[t] + (1.0f - gv) * x[t];
}

// =====================================================================
// Bidirectional LSTM recurrence. xg* already hold x@Wih^T + b.
// grid = (2 dirs, 8 batch-groups of 4), block = 512 (400 active: 4b x 100u).
// h-state in LDS, c-state in a register (one (b,u) pair per thread).
// Writes out[(b*T+t)*200 + dir*100 + u].
// =====================================================================
__global__ void lstm_scan_kernel(const float* __restrict__ xgF,
                                 const float* __restrict__ xgB,
                                 const float* __restrict__ WhhF,
                                 const float* __restrict__ WhhB,
                                 float* __restrict__ out, int T) {
    const int dir = blockIdx.x;       // 0 fwd, 1 bwd
    const int bg  = blockIdx.y;       // batch group
    const float* __restrict__ xg  = dir ? xgB  : xgF;
    const float* __restrict__ Whh = dir ? WhhB : WhhF;
    __shared__ float hsh[4 * HH];
    const int tid = threadIdx.x;
    const int bl = tid / HH, u = tid % HH;
    const int b  = bg * 4 + bl;
    float creg = 0.0f;
    if (tid < 4 * HH) hsh[tid] = 0.0f;
    __syncthreads();

    for (int s = 0; s < T; ++s) {
        const int t = dir ? (T - 1 - s) : s;
        float hn = 0.0f;
        if (tid < 4 * HH) {
            const float* hrow = hsh + bl * HH;
            const float* xrow = xg + ((size_t)b * T + t) * 400;
            float gi = xrow[u], gf = xrow[100 + u], gg = xrow[200 + u], go = xrow[300 + u];
            const float* wi = Whh + (size_t)u * HH;
            const float* wf = Whh + (size_t)(100 + u) * HH;
            const float* wg = Whh + (size_t)(200 + u) * HH;
            const float* wo = Whh + (size_t)(300 + u) * HH;
            for (int k = 0; k < HH; ++k) {
                const float hv = hrow[k];
                gi += hv * wi[k]; gf += hv * wf[k];
                gg += hv * wg[k]; go += hv * wo[k];
            }
            const float ii = 1.0f / (1.0f + __expf(-gi));
            const float ff = 1.0f / (1.0f + __expf(-gf));
            const float oo = 1.0f / (1.0f + __expf(-go));
            creg = ff * creg + ii * tanhf(gg);
            hn = oo * tanhf(creg);
        }
        __syncthreads();
        if (tid < 4 * HH) {
            hsh[tid] = hn;
            out[((size_t)b * T + t) * D2 + dir * HH + u] = hn;
        }
        __syncthreads();
    }
}

// =====================================================================
// Attention scores + row softmax + c2q, fused per (b,i) block (64 thr).
// s_j = c_i.w_c + sum_d q_jd*(w_q_d + c_id*w_cq_d) + (b_c+b_q+b_cq)
// Also emits smax[b,i] = max_j s_j (pre-softmax) for q2c.
// =====================================================================
__global__ void attn_score_kernel(const float* __restrict__ c,
                                  const float* __restrict__ q,
                                  const float* __restrict__ attn,
                                  float* __restrict__ smax,
                                  float* __restrict__ c2q) {
    __shared__ float cw2[D2], red[64], sarr[QLEN];
    const int row = blockIdx.x;          // b*512 + i
    const int b   = row >> 9;
    const int tid = threadIdx.x;         // 64
    const float* w_c  = attn;
    const float* w_q  = attn + 201;
    const float* w_cq = attn + 402;
    const float bsum  = attn[200] + attn[401] + attn[602];

    float part = 0.0f;
    for (int d = tid; d < D2; d += 64) {
        const float cv = c[(size_t)row * D2 + d];
        cw2[d] = w_q[d] + cv * w_cq[d];
        part += cv * w_c[d];
    }
    red[tid] = part; __syncthreads();
    for (int off = 32; off; off >>= 1) {
        if (tid < off) red[tid] += red[tid + off];
        __syncthreads();
    }
    const float cdot = red[0];

    if (tid < QLEN) {
        const float* qrow = q + ((size_t)b * QLEN + tid) * D2;
        float sv = cdot + bsum;
        for (int d = 0; d < D2; ++d) sv += qrow[d] * cw2[d];
        sarr[tid] = sv;
    }
    __syncthreads();
    if (tid == 0) {
        float mx = sarr[0];
        for (int j = 1; j < QLEN; ++j) mx = fmaxf(mx, sarr[j]);
        smax[row] = mx;
        float ssum = 0.0f;
        for (int j = 0; j < QLEN; ++j) { float e = __expf(sarr[j] - mx); sarr[j] = e; ssum += e; }
        const float inv = 1.0f / ssum;
        for (int j = 0; j < QLEN; ++j) sarr[j] *= inv;
    }
    __syncthreads();
    for (int d = tid; d < D2; d += 64) {
        float acc = 0.0f;
        for (int j = 0; j < QLEN; ++j)
            acc += sarr[j] * q[((size_t)b * QLEN + j) * D2 + d];
        c2q[(size_t)row * D2 + d] = acc;
    }
}

// q2c[b,d] = sum_i softmax_i(smax[b,:]) * c[b,i,d]; one block per batch.
__global__ void q2c_kernel(const float* __restrict__ smax,
                           const float* __restrict__ c,
                           float* __restrict__ q2c) {
    __shared__ float w[CLEN], red[256];
    const int b = blockIdx.x, tid = threadIdx.x;   // 256 threads
    float mx = -1e30f;
    for (int i = tid; i < CLEN; i += 256) { const float v = smax[(size_t)b*CLEN + i]; w[i] = v; mx = fmaxf(mx, v); }
    red[tid] = mx; __syncthreads();
    for (int off = 128; off; off >>= 1) { if (tid < off) red[tid] = fmaxf(red[tid], red[tid+off]); __syncthreads(); }
    mx = red[0]; __syncthreads();
    float sum = 0.0f;
    for (int i = tid; i < CLEN; i += 256) { const float e = __expf(w[i] - mx); w[i] = e; sum += e; }
    red[tid] = sum; __syncthreads();
    for (int off = 128; off; off >>= 1) { if (tid < off) red[tid] += red[tid+off]; __syncthreads(); }
    const float inv = 1.0f / red[0]; __syncthreads();
    for (int i = tid; i < CLEN; i += 256) w[i] *= inv;
    __syncthreads();
    for (int d = tid; d < D2; d += 256) {
        float acc = 0.0f;
        for (int i = 0; i < CLEN; ++i) acc += w[i] * c[((size_t)b * CLEN + i) * D2 + d];
        q2c[(size_t)b * D2 + d] = acc;
    }
}

// g = [c, c2q, c*c2q, c*q2c]  -> [16384, 800]
__global__ void build_g_kernel(const float* __restrict__ c,
                               const float* __restrict__ c2q,
                               const float* __restrict__ q2c,
                               float* __restrict__ g) {
    const int t = blockIdx.x * blockDim.x + threadIdx.x;
    if (t >= MC * D2) return;
    const int row = t / D2, d = t % D2;
    const int b = row >> 9;
    const float cv = c[t], aq = c2q[t], qc = q2c[(size_t)b * D2 + d];
    float* gr = g + (size_t)row * G8;
    gr[d] = cv; gr[D2 + d] = aq; gr[2*D2 + d] = cv * aq; gr[3*D2 + d] = cv * qc;
}

// start/end heads: dot with weight vectors, one thread per (b,i) row.
__global__ void heads_kernel(const float* __restrict__ g,
                             const float* __restrict__ m,
                             const float* __restrict__ m2,
                             const float* __restrict__ hd,
                             float* __restrict__ out) {
    const int row = blockIdx.x * blockDim.x + threadIdx.x;
    if (row >= MC) return;
    const float* Wsg = hd;          const float bsg = hd[800];
    const float* Wsm = hd + 801;    const float bsm = hd[1001];
    const float* Weg = hd + 1002;   const float beg = hd[1802];
    const float* Wem = hd + 1803;   const float bem = hd[2003];
    const float* gr  = g  + (size_t)row * G8;
    const float* mr  = m  + (size_t)row * D2;
    const float* m2r = m2 + (size_t)row * D2;
    float s = bsg + bsm, e = beg + bem;
    for (int d = 0; d < G8; ++d) { s += gr[d] * Wsg[d]; e += gr[d] * Weg[d]; }
    for (int d = 0; d < D2; ++d) { s += mr[d] * Wsm[d]; e += m2r[d] * Wem[d]; }
    out[row] = s;
    out[MC + row] = e;
}

// ---------------- host-side orchestration ----------------
static inline void launch_gemm(const float* A, const float* W, const float* bias,
                               float* C, int M, int N, int K, int act,
                               hipStream_t stream) {
    dim3 block(32, 4);
    dim3 grid((N + 15) / 16, (M / 16 + 3) / 4);
    gemm_wmma_f32<<<grid, block, 0, stream>>>(A, W, bias, C, M, N, K, act);
}

extern "C" void kernel_launch(void* const* d_in, const int* in_sizes, int n_in,
                              void* d_out, int out_size, void* d_ws, size_t ws_size,
                              hipStream_t stream) {
    // ---- inputs (setup_inputs order; nested dicts concatenated in insertion order)
    const int*   c_char   = (const int*)  d_in[0];
    const int*   q_char   = (const int*)  d_in[1];
    const int*   c_word   = (const int*)  d_in[2];
    const int*   q_word   = (const int*)  d_in[3];
    const float* char_emb = (const float*)d_in[4];
    const float* conv_W   = (const float*)d_in[5];
    const float* conv_b   = (const float*)d_in[6];
    const float* word_emb = (const float*)d_in[7];
    const float* hw       = (const float*)d_in[8];   // Wl[2,200,200], bl[2,200], Wg, bg
    const float* ctx      = (const float*)d_in[9];   // lstm_p(din=200)
    const float* mo1      = (const float*)d_in[10];  // lstm_p(din=800)
    const float* mo2      = (const float*)d_in[11];  // lstm_p(din=200)
    const float* olp      = (const float*)d_in[12];  // lstm_p(din=200)
    const float* attn     = (const float*)d_in[13];
    const float* hd       = (const float*)d_in[14];
    float* out = (float*)d_out;

    // lstm_p offsets (din=200): Wih_f 0, Whh_f 80000, b_f 120000,
    //                           Wih_b 120400, Whh_b 200400, b_b 240400
    // lstm_p offsets (din=800): Wih_f 0, Whh_f 320000, b_f 360000,
    //                           Wih_b 360400, Whh_b 680400, b_b 720400

    // ---- workspace layout (floats)
    float* ws = (float*)d_ws;
    float* g    = ws;                       // [16384, 800]
    float* xgf  = ws + 13107200;            // [16384, 400]
    float* xgb  = ws + 19660800;            // [16384, 400]
    float* hxc  = ws + 26214400;            // [16384, 200] highway input/output (ctx)
    float* cout = ws + 29491200;            // [16384, 200] ctx bilstm out
    float* t1   = ws + 32768000;            // [16384, 200] highway h / c2q
    float* t2   = ws + 36044800;            // [16384, 200] highway gate
    float* m1b  = ws + 39321600;            // [16384, 200] mod_lstm1 out
    float* mb   = ws + 42598400;            // [16384, 200] mod_lstm2 out (m)
    float* m2b  = ws + 45875200;            // [16384, 200] out_lstm out (m2)
    float* hxq  = ws + 49152000;            // [1536, 200]  highway (query)
    float* qout = ws + 49459200;            // [1536, 200]  q bilstm out
    float* smax = ws + 49766400;            // [16384]
    float* q2c  = ws + 49782784;            // [32, 200]

    // ---- 1. char CNN + word gather -> highway inputs
    char_conv_kernel<<<MC, 128, 0, stream>>>(c_char, char_emb, conv_W, conv_b, hxc);
    char_conv_kernel<<<MQ, 128, 0, stream>>>(q_char, char_emb, conv_W, conv_b, hxq);
    word_gather_kernel<<<(MC*100 + 255)/256, 256, 0, stream>>>(c_word, word_emb, hxc, MC);
    word_gather_kernel<<<(MQ*100 + 255)/256, 256, 0, stream>>>(q_word, word_emb, hxq, MQ);

    // ---- 2. highway (2 layers) for ctx and query
    for (int pass = 0; pass < 2; ++pass) {
        float* x = pass ? hxq : hxc;
        const int M = pass ? MQ : MC;
        for (int il = 0; il < 2; ++il) {
            launch_gemm(x, hw + il*40000,         hw + 80000  + il*200, t1, M, D2, D2, 1, stream);
            launch_gemm(x, hw + 80400 + il*40000, hw + 160400 + il*200, t2, M, D2, D2, 2, stream);
            highway_combine_kernel<<<(M*D2 + 255)/256, 256, 0, stream>>>(x, t1, t2, M*D2);
        }
    }

    // ---- 3. context bilstm (input 200)
    launch_gemm(hxc, ctx,          ctx + 120000, xgf, MC, 400, D2, 0, stream);
    launch_gemm(hxc, ctx + 120400, ctx + 240400, xgb, MC, 400, D2, 0, stream);
    lstm_scan_kernel<<<dim3(2, 8), 512, 0, stream>>>(xgf, xgb, ctx + 80000, ctx + 200400, cout, CLEN);

    // ---- 4. query bilstm (same weights)
    launch_gemm(hxq, ctx,          ctx + 120000, xgf, MQ, 400, D2, 0, stream);
    launch_gemm(hxq, ctx + 120400, ctx + 240400, xgb, MQ, 400, D2, 0, stream);
    lstm_scan_kernel<<<dim3(2, 8), 512, 0, stream>>>(xgf, xgb, ctx + 80000, ctx + 200400, qout, QLEN);

    // ---- 5. attention flow -> g [16384, 800]
    attn_score_kernel<<<MC, 64, 0, stream>>>(cout, qout, attn, smax, t1 /*c2q*/);
    q2c_kernel<<<BQ, 256, 0, stream>>>(smax, cout, q2c);
    build_g_kernel<<<(MC*D2 + 255)/256, 256, 0, stream>>>(cout, t1, q2c, g);

    // ---- 6. modeling bilstm 1 (input 800)
    launch_gemm(g, mo1,          mo1 + 360000, xgf, MC, 400, G8, 0, stream);
    launch_gemm(g, mo1 + 360400, mo1 + 720400, xgb, MC, 400, G8, 0, stream);
    lstm_scan_kernel<<<dim3(2, 8), 512, 0, stream>>>(xgf, xgb, mo1 + 320000, mo1 + 680400, m1b, CLEN);

    // ---- 7. modeling bilstm 2 (input 200) -> m
    launch_gemm(m1b, mo2,          mo2 + 120000, xgf, MC, 400, D2, 0, stream);
    launch_gemm(m1b, mo2 + 120400, mo2 + 240400, xgb, MC, 400, D2, 0, stream);
    lstm_scan_kernel<<<dim3(2, 8), 512, 0, stream>>>(xgf, xgb, mo2 + 80000, mo2 + 200400, mb, CLEN);

    // ---- 8. output bilstm (input 200) -> m2
    launch_gemm(mb, olp,          olp + 120000, xgf, MC, 400, D2, 0, stream);
    launch_gemm(mb, olp + 120400, olp + 240400, xgb, MC, 400, D2, 0, stream);
    lstm_scan_kernel<<<dim3(2, 8), 512, 0, stream>>>(xgf, xgb, olp + 80000, olp + 200400, m2b, CLEN);

    // ---- 9. start/end heads
    heads_kernel<<<(MC + 255)/256, 256, 0, stream>>>(g, mb, m2b, hd, out);

    (void)in_sizes; (void)n_in; (void)out_size; (void)ws_size;
}